// GCNLayer_44650480009877
// MI455X (gfx1250) — compile-verified
//
#include <hip/hip_runtime.h>

#define N_NODES 100000
#define N_EDGES 640000
#define DIM 128
#define BN_EPS 1e-5f

typedef __attribute__((ext_vector_type(2))) float v2f;
typedef __attribute__((ext_vector_type(8))) float v8f;

// ---------------------------------------------------------------------------
// Kernel 1: zero the accumulator (d_out) and the stats workspace.
// ---------------------------------------------------------------------------
__global__ void zero_kernel(float4* __restrict__ out4, float* __restrict__ ws) {
    int i = blockIdx.x * blockDim.x + threadIdx.x;
    if (i < (N_NODES * DIM / 4)) out4[i] = make_float4(0.f, 0.f, 0.f, 0.f);
    if (i < 256) ws[i] = 0.f;  // sum[128] + sumsq[128]
}

// ---------------------------------------------------------------------------
// Kernel 2: edge scatter-add.  One wave32 per edge; lane l handles dims
// [4l, 4l+4) as a float4 load + 4 hardware fp32 atomics (global_atomic_add_f32).
// ---------------------------------------------------------------------------
__global__ void scatter_kernel(const float* __restrict__ feat,
                               const float* __restrict__ ew,
                               const int*   __restrict__ src,
                               const int*   __restrict__ dst,
                               float*       __restrict__ h) {
    int gid  = blockIdx.x * blockDim.x + threadIdx.x;
    int e    = gid >> 5;
    int lane = gid & 31;
    if (e >= N_EDGES) return;
    int   s = src[e];
    int   d = dst[e];
    float w = ew[e];
    const float4 f = *reinterpret_cast<const float4*>(feat + (size_t)s * DIM + lane * 4);
    float* hp = h + (size_t)d * DIM + lane * 4;
    unsafeAtomicAdd(hp + 0, w * f.x);
    unsafeAtomicAdd(hp + 1, w * f.y);
    unsafeAtomicAdd(hp + 2, w * f.z);
    unsafeAtomicAdd(hp + 3, w * f.w);
}

// ---------------------------------------------------------------------------
// Kernel 3: Y = relu(H @ W^T + b) in place over d_out, via fp32 WMMA
// (v_wmma_f32_16x16x4_f32), fused with per-channel sum/sumsq for BatchNorm.
// Block = 256 threads = 8 waves; block owns a 16-row M tile (staged in LDS),
// wave w owns the 16-column N tile n0 = 16*w (DIM = 128 = 8 * 16).
//
// A-fragment (16x4 f32, 2 VGPRs): lanes 0-15 hold K=k0,k0+1 of row M=lane;
// lanes 16-31 hold K=k0+2,k0+3 of row M=lane-16.
// B-fragment (4x16 f32, 2 VGPRs): lane holds column N=lane%16,
// rows K=k0+2*(lane/16) and K=k0+2*(lane/16)+1;  B[k][n] = W[n][k].
// D-fragment: VGPR v = row (lane<16 ? v : v+8), column N=lane%16 — every lane
// holds 8 values of ONE channel, ideal for the BN reduction.
// ---------------------------------------------------------------------------
__launch_bounds__(256)
__global__ void gemm_relu_stats_kernel(float* __restrict__ hy,      // in: H, out: Y (in place)
                                       const float* __restrict__ W, // [128 out][128 in]
                                       const float* __restrict__ bias,
                                       float* __restrict__ sum,
                                       float* __restrict__ sumsq) {
    __shared__ float tile[16 * DIM];  // 8 KB H tile

    const int m0 = blockIdx.x * 16;

    // Stage the 16x128 H tile: 512 float4s, 2 per thread.
    {
        const float4* s4 = reinterpret_cast<const float4*>(hy + (size_t)m0 * DIM);
        float4*       d4 = reinterpret_cast<float4*>(tile);
        d4[threadIdx.x]       = s4[threadIdx.x];
        d4[threadIdx.x + 256] = s4[threadIdx.x + 256];
    }
    __syncthreads();

    const int wave = threadIdx.x >> 5;
    const int lane = threadIdx.x & 31;
    const int nl   = lane & 15;
    const int half = lane >> 4;
    const int n    = wave * 16 + nl;     // this lane's output channel

    const float* wrow = W + (size_t)n * DIM;

    v8f acc = {};
#pragma unroll
    for (int k0 = 0; k0 < DIM; k0 += 4) {
        const int kk = k0 + 2 * half;
        v2f a = *reinterpret_cast<const v2f*>(&tile[nl * DIM + kk]);   // ds_load_b64
        v2f b = *reinterpret_cast<const v2f*>(wrow + kk);              // global_load_b64
        acc = __builtin_amdgcn_wmma_f32_16x16x4_f32(
            /*neg_a=*/false, a, /*neg_b=*/false, b,
            /*c_mod=*/(short)0, acc, /*reuse_a=*/false, /*reuse_b=*/false);
    }

    // Epilogue: bias + ReLU, store Y, accumulate channel stats.
    const float bn = bias[n];
    float ls = 0.f, lq = 0.f;
    float* yb = hy + (size_t)m0 * DIM + n;
#pragma unroll
    for (int v = 0; v < 8; ++v) {
        float val = fmaxf(acc[v] + bn, 0.f);
        const int m = half ? (v + 8) : v;
        yb[(size_t)m * DIM] = val;
        ls += val;
        lq += val * val;
    }
    // Lanes L and L+16 hold the same channel n: fold the pair, one atomic each.
    ls += __shfl_xor(ls, 16);
    lq += __shfl_xor(lq, 16);
    if (half == 0) {
        unsafeAtomicAdd(sum + n, ls);
        unsafeAtomicAdd(sumsq + n, lq);
    }
}

// ---------------------------------------------------------------------------
// Kernel 4: fold sum/sumsq into per-channel scale/shift.
// ---------------------------------------------------------------------------
__global__ void bn_finalize_kernel(const float* __restrict__ sum,
                                   const float* __restrict__ sumsq,
                                   const float* __restrict__ gamma,
                                   const float* __restrict__ beta,
                                   float* __restrict__ scale,
                                   float* __restrict__ shift) {
    int n = threadIdx.x;  // 128 threads
    const float inv  = 1.0f / (float)N_NODES;
    float mean = sum[n] * inv;
    float var  = fmaf(-mean, mean, sumsq[n] * inv);  // E[x^2] - mean^2 (biased)
    float sc   = gamma[n] * rsqrtf(var + BN_EPS);
    scale[n] = sc;
    shift[n] = fmaf(-mean, sc, beta[n]);
}

// ---------------------------------------------------------------------------
// Kernel 5: apply BN in place, vectorized float4.
// ---------------------------------------------------------------------------
__global__ void bn_apply_kernel(float4* __restrict__ y4,
                                const float4* __restrict__ scale4,
                                const float4* __restrict__ shift4) {
    int i = blockIdx.x * blockDim.x + threadIdx.x;
    if (i >= (N_NODES * DIM / 4)) return;
    const int c = i & 31;  // 128/4 channel-quads per row
    float4 v = y4[i];
    const float4 s = scale4[c];
    const float4 t = shift4[c];
    v.x = fmaf(v.x, s.x, t.x);
    v.y = fmaf(v.y, s.y, t.y);
    v.z = fmaf(v.z, s.z, t.z);
    v.w = fmaf(v.w, s.w, t.w);
    y4[i] = v;
}

// ---------------------------------------------------------------------------
extern "C" void kernel_launch(void* const* d_in, const int* in_sizes, int n_in,
                              void* d_out, int out_size, void* d_ws, size_t ws_size,
                              hipStream_t stream) {
    const float* feature = (const float*)d_in[0];
    const float* ew      = (const float*)d_in[1];
    const int*   src     = (const int*)d_in[2];
    const int*   dst     = (const int*)d_in[3];
    const float* W       = (const float*)d_in[4];
    const float* b       = (const float*)d_in[5];
    const float* gamma   = (const float*)d_in[6];
    const float* beta    = (const float*)d_in[7];

    float* out = (float*)d_out;
    float* ws  = (float*)d_ws;
    float* sum   = ws;
    float* sumsq = ws + 128;
    float* scale = ws + 256;
    float* shift = ws + 384;

    const int total4 = N_NODES * DIM / 4;  // 3.2M float4s

    zero_kernel<<<(total4 + 255) / 256, 256, 0, stream>>>((float4*)out, ws);

    const long sthreads = (long)N_EDGES * 32;
    scatter_kernel<<<(int)((sthreads + 255) / 256), 256, 0, stream>>>(
        feature, ew, src, dst, out);

    gemm_relu_stats_kernel<<<N_NODES / 16, 256, 0, stream>>>(out, W, b, sum, sumsq);

    bn_finalize_kernel<<<1, 128, 0, stream>>>(sum, sumsq, gamma, beta, scale, shift);

    bn_apply_kernel<<<(total4 + 255) / 256, 256, 0, stream>>>(
        (float4*)out, (const float4*)scale, (const float4*)shift);
}